// GaussianNoise_61426622267935
// MI455X (gfx1250) — compile-verified
//
#include <hip/hip_runtime.h>

// out = clip(x + noise, 0, 1), fp32 elementwise.
// HBM-streaming kernel: 12 B/elem, ~604 MB total -> ~26 us floor at 23.3 TB/s.
// Path: global_load_b128 (TH=NT) x2 -> v_add_f32 with clamp modifier ->
//       global_store_b128 (TH=NT). One VALU op per element, zero loop overhead.

typedef float v4f __attribute__((ext_vector_type(4)));

__device__ __forceinline__ float clamp01(float v) {
    // fmin(fmax(v,0),1) folds into the AMDGPU 'clamp' output modifier on the
    // instruction that produced v (v_add_f32 ... clamp).
    return __builtin_fminf(__builtin_fmaxf(v, 0.0f), 1.0f);
}

__global__ __launch_bounds__(256) void GaussianNoise_v4_kernel(
    const v4f* __restrict__ x,
    const v4f* __restrict__ nz,
    v4f* __restrict__ out,
    int n4)
{
    const int i = blockIdx.x * blockDim.x + threadIdx.x;
    if (i < n4) {
        // Non-temporal: streamed once, keep the 192MB L2 clean (TH=NT).
        v4f a = __builtin_nontemporal_load(x + i);
        v4f b = __builtin_nontemporal_load(nz + i);
        v4f r;
        r.x = clamp01(a.x + b.x);
        r.y = clamp01(a.y + b.y);
        r.z = clamp01(a.z + b.z);
        r.w = clamp01(a.w + b.w);
        __builtin_nontemporal_store(r, out + i);
    }
}

// Scalar tail for element counts not divisible by 4 (not hit for 64*3*512*512,
// but keeps the launch fully general).
__global__ __launch_bounds__(256) void GaussianNoise_tail_kernel(
    const float* __restrict__ x,
    const float* __restrict__ nz,
    float* __restrict__ out,
    int start, int n)
{
    const int i = start + blockIdx.x * blockDim.x + threadIdx.x;
    if (i < n) {
        out[i] = clamp01(x[i] + nz[i]);
    }
}

extern "C" void kernel_launch(void* const* d_in, const int* in_sizes, int n_in,
                              void* d_out, int out_size, void* d_ws, size_t ws_size,
                              hipStream_t stream) {
    const float* x   = (const float*)d_in[0];
    const float* nz  = (const float*)d_in[1];
    float*       out = (float*)d_out;

    const int n  = in_sizes[0];   // 50,331,648 for the reference shapes
    const int n4 = n >> 2;        // float4 count (hipMalloc base is 256B-aligned)

    if (n4 > 0) {
        const int block = 256;    // 8 wave32s per block
        const unsigned grid = (unsigned)((n4 + block - 1) / block);  // 49,152 blocks
        GaussianNoise_v4_kernel<<<grid, block, 0, stream>>>(
            (const v4f*)x, (const v4f*)nz, (v4f*)out, n4);
    }

    const int done = n4 << 2;
    const int rem  = n - done;
    if (rem > 0) {
        const int block = 256;
        const unsigned grid = (unsigned)((rem + block - 1) / block);
        GaussianNoise_tail_kernel<<<grid, block, 0, stream>>>(x, nz, out, done, n);
    }
}